// UnifiedResidueGeometry_15547781611608
// MI455X (gfx1250) — compile-verified
//
#include <hip/hip_runtime.h>

#define NB   2
#define LL   2048
#define DD   96
#define HH   4
#define HK   64      // H*DK
#define OUTD 96
#define INF_ 100000.0f
#define EPS_DIR 1e-10f
#define EPS_LN  1e-5f

typedef float v2f __attribute__((ext_vector_type(2)));
typedef float v8f __attribute__((ext_vector_type(8)));

static __device__ __forceinline__ v8f wmma_f32(v2f a, v2f b, v8f c) {
    // D(16x16,f32) = A(16x4,f32) x B(4x16,f32) + C
    return __builtin_amdgcn_wmma_f32_16x16x4_f32(false, a, false, b, (short)0, c, false, false);
}

// ---------------------------------------------------------------------------
// Kernel 1: Q/K/V projections. grid = (L/16 row tiles, N, 12 = {q,k,v} x 4 col tiles)
// ---------------------------------------------------------------------------
__global__ __launch_bounds__(32)
void qkv_proj_kernel(const float* __restrict__ x,
                     const float* __restrict__ Wq,
                     const float* __restrict__ Wk,
                     const float* __restrict__ Wv,
                     float* __restrict__ qo, float* __restrict__ ko, float* __restrict__ vo)
{
    const int lane = threadIdx.x & 31;
    const int half = lane >> 4, ln = lane & 15;
    const int rowbase = blockIdx.x * 16;
    const int b  = blockIdx.y;
    const int mat = blockIdx.z >> 2, ct = blockIdx.z & 3;
    const float* W  = (mat == 0) ? Wq : (mat == 1) ? Wk : Wv;
    float*       out = (mat == 0) ? qo : (mat == 1) ? ko : vo;

    const float* xrow = x + ((size_t)b * LL + rowbase + ln) * DD;
    v8f acc = {};
#pragma unroll
    for (int kc = 0; kc < DD / 4; ++kc) {
        const int k0 = kc * 4 + 2 * half;           // A layout: vgpr0 -> K=0|2, vgpr1 -> K=1|3
        v2f a, bm;
        a[0]  = xrow[k0 + 0];
        a[1]  = xrow[k0 + 1];
        bm[0] = W[(k0 + 0) * HK + ct * 16 + ln];    // B layout mirrors A along K
        bm[1] = W[(k0 + 1) * HK + ct * 16 + ln];
        acc = wmma_f32(a, bm, acc);
    }
    float* orow = out + ((size_t)b * LL + rowbase) * HK + ct * 16 + ln;
#pragma unroll
    for (int r = 0; r < 8; ++r)
        orow[(size_t)(r + 8 * half) * HK] = acc[r]; // C layout: vgpr r -> row r | r+8
}

// ---------------------------------------------------------------------------
// Kernel 2: flash attention + fused geometry accumulation.
// grid = (L/16 query tiles, H, N), one wave per block.
// ---------------------------------------------------------------------------
__global__ __launch_bounds__(32)
void attn_kernel(const float* __restrict__ q, const float* __restrict__ k,
                 const float* __restrict__ v,
                 const float* __restrict__ pos_CA, const float* __restrict__ pos_CB,
                 const unsigned char* __restrict__ mask,
                 float* __restrict__ fn,   // (N,L,H*16) feat_node
                 float* __restrict__ apb)  // (N,L,H*3) atom_pos_bias
{
    __shared__ float pt[16][16];
    const int lane = threadIdx.x & 31;
    const int half = lane >> 4, ln = lane & 15;
    const int qbase = blockIdx.x * 16;
    const int h = blockIdx.y, b = blockIdx.z;

    // Q tile in A-fragment layout (4 fragments x 2 VGPRs)
    float aq[8];
    const float* qrow = q + ((size_t)b * LL + qbase + ln) * HK + h * 16;
#pragma unroll
    for (int f = 0; f < 4; ++f) {
        const int k0 = 4 * f + 2 * half;
        aq[2 * f + 0] = qrow[k0 + 0];
        aq[2 * f + 1] = qrow[k0 + 1];
    }

    float m8[8], l8[8];
    v8f accv = {}, acce = {};
#pragma unroll
    for (int r = 0; r < 8; ++r) { m8[r] = -1e30f; l8[r] = 0.f; }

    for (int kt = 0; kt < LL / 16; ++kt) {
        const int kbase = kt * 16;
        // ---- S = Q Kt : 4 chained f32 WMMAs ----
        const float* krow = k + ((size_t)b * LL + kbase + ln) * HK + h * 16;
        v8f s = {};
#pragma unroll
        for (int f = 0; f < 4; ++f) {
            const int k0 = 4 * f + 2 * half;
            v2f a  = { aq[2 * f], aq[2 * f + 1] };
            v2f bm = { krow[k0 + 0], krow[k0 + 1] };
            s = wmma_f32(a, bm, s);
        }
        // key-column mask (C layout: this lane owns key kbase+ln)
        const float maskadd = mask[(size_t)b * LL + kbase + ln] ? 0.f : -INF_;

        // ---- online softmax (per-row state matches C-layout rows) ----
        float p[8];
#pragma unroll
        for (int r = 0; r < 8; ++r) {
            float sv = s[r] + maskadd;
            float rm = sv;                               // rowmax over the 16 lanes of this half
            rm = fmaxf(rm, __shfl_xor(rm, 1));
            rm = fmaxf(rm, __shfl_xor(rm, 2));
            rm = fmaxf(rm, __shfl_xor(rm, 4));
            rm = fmaxf(rm, __shfl_xor(rm, 8));
            const float nm = fmaxf(m8[r], rm);
            const float sc = __expf(m8[r] - nm);
            const float pv = __expf(sv - nm);
            float rs = pv;
            rs += __shfl_xor(rs, 1);
            rs += __shfl_xor(rs, 2);
            rs += __shfl_xor(rs, 4);
            rs += __shfl_xor(rs, 8);
            l8[r] = l8[r] * sc + rs;
            m8[r] = nm;
            accv[r] *= sc;
            acce[r] *= sc;
            p[r] = pv;
        }

        // ---- relayout P: C-layout -> A-fragments through LDS ----
        __syncthreads();
#pragma unroll
        for (int r = 0; r < 8; ++r) pt[r + 8 * half][ln] = p[r];
        __syncthreads();
        float pa[8];
#pragma unroll
        for (int f = 0; f < 4; ++f) {
            pa[2 * f + 0] = pt[ln][4 * f + 2 * half + 0];
            pa[2 * f + 1] = pt[ln][4 * f + 2 * half + 1];
        }

        // ---- accv += P V ; acce += P [CB.x CB.y CB.z 1 0...] ----
#pragma unroll
        for (int f = 0; f < 4; ++f) {
            const int krel = 4 * f + 2 * half;           // B layout: vgpr holds K-rows krel, krel+1
            v2f a = { pa[2 * f], pa[2 * f + 1] };
            v2f bv, be;
            bv[0] = v[((size_t)b * LL + kbase + krel + 0) * HK + h * 16 + ln];
            bv[1] = v[((size_t)b * LL + kbase + krel + 1) * HK + h * 16 + ln];
            accv = wmma_f32(a, bv, accv);
            const int key0 = kbase + krel;
            be[0] = (ln < 3) ? pos_CB[((size_t)b * LL + key0 + 0) * 3 + ln] : (ln == 3 ? 1.f : 0.f);
            be[1] = (ln < 3) ? pos_CB[((size_t)b * LL + key0 + 1) * 3 + ln] : (ln == 3 ? 1.f : 0.f);
            acce = wmma_f32(a, be, acce);
        }
    }

    // ---- finalize: normalize, subtract CA * (sum alpha), write out ----
#pragma unroll
    for (int r = 0; r < 8; ++r) {
        const int row = qbase + r + 8 * half;
        const float invl = 1.0f / l8[r];
        fn[((size_t)b * LL + row) * HK + h * 16 + ln] = accv[r] * invl;
        const float oe = acce[r] * invl;
        const float asum = __shfl(oe, 16 * half + 3);   // col 3 holds sum(alpha)
        if (ln < 3) {
            const float val = oe - pos_CA[((size_t)b * LL + row) * 3 + ln] * asum;
            apb[((size_t)b * LL + row) * (HH * 3) + h * 3 + ln] = val;
        }
    }
}

// ---------------------------------------------------------------------------
// Kernel 3: geometry features + output projection + residual + LayerNorm.
// grid = (L, N), block = 96 threads (one per output channel).
// ---------------------------------------------------------------------------
__global__ __launch_bounds__(96)
void out_kernel(const float* __restrict__ x, const float* __restrict__ frame,
                const unsigned char* __restrict__ mask,
                const float* __restrict__ Wo, const float* __restrict__ bo,
                const float* __restrict__ gamma, const float* __restrict__ beta,
                const float* __restrict__ fn, const float* __restrict__ apb,
                float* __restrict__ out)
{
    __shared__ float feat[92];
    __shared__ float red[128];
    const int t = threadIdx.x;
    const size_t bl = (size_t)blockIdx.y * LL + blockIdx.x;

    if (t < 64) {                                   // feat_node
        feat[t] = fn[bl * HK + t];
    } else if (t < 76) {                            // feat_points = frame @ apb
        const int idx = t - 64, h = idx / 3, i = idx % 3;
        float s = 0.f;
        for (int j = 0; j < 3; ++j) s += frame[bl * 9 + i * 3 + j] * apb[bl * 12 + h * 3 + j];
        feat[t] = s;
    } else if (t < 80) {                            // feat_distance = |apb|
        const int h = t - 76;
        float s = 0.f;
        for (int j = 0; j < 3; ++j) { const float a = apb[bl * 12 + h * 3 + j]; s += a * a; }
        feat[t] = sqrtf(s);
    }
    __syncthreads();
    if (t >= 80 && t < 92) {                        // feat_direction = fp/(|fp|+eps)
        const int idx = t - 80, h = idx / 3;
        const float fx = feat[64 + h * 3 + 0], fy = feat[64 + h * 3 + 1], fz = feat[64 + h * 3 + 2];
        const float nrm = sqrtf(fx * fx + fy * fy + fz * fz);
        feat[t] = feat[64 + idx] / (nrm + EPS_DIR);
    }
    __syncthreads();

    float o = bo[t];
    for (int r = 0; r < 92; ++r) o += feat[r] * Wo[r * OUTD + t];
    const float hv = x[bl * OUTD + t] + (mask[bl] ? o : 0.f);

    red[t] = hv; if (t < 32) red[96 + t] = 0.f;
    __syncthreads();
    for (int s2 = 64; s2 > 0; s2 >>= 1) { if (t < s2) red[t] += red[t + s2]; __syncthreads(); }
    const float mu = red[0] * (1.0f / OUTD);
    __syncthreads();
    const float d = hv - mu;
    red[t] = d * d; if (t < 32) red[96 + t] = 0.f;
    __syncthreads();
    for (int s2 = 64; s2 > 0; s2 >>= 1) { if (t < s2) red[t] += red[t + s2]; __syncthreads(); }
    const float var = red[0] * (1.0f / OUTD);
    out[bl * OUTD + t] = d * rsqrtf(var + EPS_LN) * gamma[t] + beta[t];
}

// ---------------------------------------------------------------------------
extern "C" void kernel_launch(void* const* d_in, const int* in_sizes, int n_in,
                              void* d_out, int out_size, void* d_ws, size_t ws_size,
                              hipStream_t stream)
{
    const float*         x      = (const float*)d_in[0];
    const float*         pos_CA = (const float*)d_in[1];
    const float*         pos_CB = (const float*)d_in[2];
    const float*         frame  = (const float*)d_in[3];
    const unsigned char* mask   = (const unsigned char*)d_in[4];
    const float*         Wq     = (const float*)d_in[5];
    const float*         Wk     = (const float*)d_in[6];
    const float*         Wv     = (const float*)d_in[7];
    const float*         Wo     = (const float*)d_in[8];
    const float*         bo     = (const float*)d_in[9];
    const float*         gamma  = (const float*)d_in[10];
    const float*         beta   = (const float*)d_in[11];

    float* ws   = (float*)d_ws;
    const size_t NLHK = (size_t)NB * LL * HK;       // 262144 floats
    float* qb   = ws;
    float* kb   = ws + NLHK;
    float* vb   = ws + 2 * NLHK;
    float* fnb  = ws + 3 * NLHK;
    float* apbb = ws + 4 * NLHK;                    // N*L*12 floats
    float* outp = (float*)d_out;

    dim3 g1(LL / 16, NB, 12);
    qkv_proj_kernel<<<g1, 32, 0, stream>>>(x, Wq, Wk, Wv, qb, kb, vb);

    dim3 g2(LL / 16, HH, NB);
    attn_kernel<<<g2, 32, 0, stream>>>(qb, kb, vb, pos_CA, pos_CB, mask, fnb, apbb);

    dim3 g3(LL, NB);
    out_kernel<<<g3, 96, 0, stream>>>(x, frame, mask, Wo, bo, gamma, beta, fnb, apbb, outp);
}